// GCN_42975442764292
// MI455X (gfx1250) — compile-verified
//
#include <hip/hip_runtime.h>
#include <hip/hip_bf16.h>

// Problem constants (match reference)
#define NN     10000
#define EE     160000
#define BB     2
#define CIN    16
#define CHID   32
#define TT     8
#define NCLASS 23
#define KK     (NN - NCLASS + 1)   // 9978
#define EPSLN  1e-5f
#define JD     (BB * TT * CIN)     // 256 channels per node (layer-1 slab)
#define JD2    (BB * TT * CHID)    // 512 channels per node (layer-2 slab)

typedef __attribute__((ext_vector_type(2))) float v2f;
typedef __attribute__((ext_vector_type(8))) float v8f;

// ---------------------------------------------------------------- utilities
__global__ void zero_f32(float* p, int n) {
    int i = blockIdx.x * blockDim.x + threadIdx.x;
    if (i < n) p[i] = 0.0f;
}

__device__ __forceinline__ float block_reduce_sum(float v, float* sm) {
    int tid = threadIdx.x;
    sm[tid] = v;
    __syncthreads();
    for (int s = blockDim.x >> 1; s > 0; s >>= 1) {
        if (tid < s) sm[tid] += sm[tid + s];
        __syncthreads();
    }
    float r = sm[0];
    __syncthreads();
    return r;
}

// ------------------------------------------ 0) features -> node-major slabs
// xT[n][j] with j = b*128 + t*16 + c,  from feat[((b*16+c)*8+t)*N + n].
// LDS-tiled 16x16 transpose: reads coalesced along n, writes along j.
__global__ void transpose_feat(const float* __restrict__ feat,
                               float* __restrict__ xT) {
    __shared__ float tile[16][17];
    int nt = blockIdx.x;                 // 625 node tiles
    int jt = blockIdx.y;                 // 16 channel tiles
    int tx = threadIdx.x & 15, ty = threadIdx.x >> 4;

    int j = jt * 16 + ty;                // output channel row
    int b = j >> 7, t = (j >> 4) & 7, c = j & 15;
    int rf = (b * CIN + c) * TT + t;     // feature row for this j
    tile[ty][tx] = feat[(size_t)rf * NN + nt * 16 + tx];
    __syncthreads();
    xT[(size_t)(nt * 16 + ty) * JD + jt * 16 + tx] = tile[tx][ty];
}

// ------------------------------------------------- 1) edge scatter (layer 1)
// agg1[dst-slab] += xT[src-slab]; slab = 256 floats = 64 float4.
// 256-thread block handles 4 edges (64 lanes each, b128 loads).
__global__ void scatter_agg1(const float4* __restrict__ xT4,
                             const int* __restrict__ src,
                             const int* __restrict__ dst,
                             float* __restrict__ agg1) {
    int e = blockIdx.x * 4 + (threadIdx.x >> 6);
    int l = threadIdx.x & 63;
    int s = src[e], d = dst[e];
    float4 v = xT4[(size_t)s * 64 + l];
    float* dp = agg1 + (size_t)d * JD + l * 4;
    atomicAdd(dp + 0, v.x);
    atomicAdd(dp + 1, v.y);
    atomicAdd(dp + 2, v.z);
    atomicAdd(dp + 3, v.w);
}

// ------------------------------------------------- 2) h1 = relu(agg1@W1+b1)
// WMMA f32 16x16x4: one wave per 16-row tile, K=16 in 4 steps,
// two 16x16 accumulator tiles cover the 32 output channels.
__global__ void h1_wmma(const float* __restrict__ agg1,
                        const float* __restrict__ W1,
                        const float* __restrict__ b1,
                        float* __restrict__ h1) {
    int wave    = blockIdx.x * (blockDim.x >> 5) + (threadIdx.x >> 5);
    int lane    = threadIdx.x & 31;
    int rowbase = wave * 16;             // 10000 tiles cover all 160000 rows
    int m    = lane & 15;
    int half = lane >> 4;

    const float* arow = agg1 + (size_t)(rowbase + m) * CIN;

    v8f c0 = {};
    v8f c1 = {};
#pragma unroll
    for (int kk = 0; kk < 4; ++kk) {
        int k0 = kk * 4 + 2 * half;
        v2f a;                           // A(m, k0), A(m, k0+1)
        a[0] = arow[k0];
        a[1] = arow[k0 + 1];
        v2f bb0, bb1;                    // B(k, n) = W1[k*CHID + n]
        bb0[0] = W1[k0 * CHID + m];
        bb0[1] = W1[(k0 + 1) * CHID + m];
        bb1[0] = W1[k0 * CHID + 16 + m];
        bb1[1] = W1[(k0 + 1) * CHID + 16 + m];
        c0 = __builtin_amdgcn_wmma_f32_16x16x4_f32(false, a, false, bb0,
                                                   (short)0, c0, false, false);
        c1 = __builtin_amdgcn_wmma_f32_16x16x4_f32(false, a, false, bb1,
                                                   (short)0, c1, false, false);
    }

    float bias0 = b1[m];
    float bias1 = b1[16 + m];
#pragma unroll
    for (int r = 0; r < 8; ++r) {
        int M = r + 8 * half;
        float v0 = c0[r] + bias0;
        float v1 = c1[r] + bias1;
        v0 = v0 > 0.0f ? v0 : 0.0f;
        v1 = v1 > 0.0f ? v1 : 0.0f;
        float* orow = h1 + (size_t)(rowbase + M) * CHID;
        orow[m]      = v0;
        orow[16 + m] = v1;
    }
}

// ------------------------------------------------- 3) edge scatter (layer 2)
// agg2[dst-slab] += h1[src-slab]; slab = 512 floats = 128 float4.
// 256-thread block handles 2 edges (128 lanes each, b128 loads).
__global__ void scatter_agg2(const float4* __restrict__ h14,
                             const int* __restrict__ src,
                             const int* __restrict__ dst,
                             float* __restrict__ agg2) {
    int e = blockIdx.x * 2 + (threadIdx.x >> 7);
    int l = threadIdx.x & 127;
    int s = src[e], d = dst[e];
    float4 v = h14[(size_t)s * 128 + l];
    float* dp = agg2 + (size_t)d * JD2 + l * 4;
    atomicAdd(dp + 0, v.x);
    atomicAdd(dp + 1, v.y);
    atomicAdd(dp + 2, v.z);
    atomicAdd(dp + 3, v.w);
}

// --------------------------------- 4) h2 = agg2@W2+b2 fused with tconv1 (T)
// z[b,n] = sum_{t,o} agg2[n,b,t,o]*W2[o]*tw[t] + b2*sum_t tw[t] + tb
// One wave per (n,b); lane = output channel o.
__global__ void h2_tconv1(const float* __restrict__ agg2,
                          const float* __restrict__ W2,
                          const float* __restrict__ b2,
                          const float* __restrict__ tw,
                          const float* __restrict__ tb,
                          float* __restrict__ z) {
    int gw   = blockIdx.x * (blockDim.x >> 5) + (threadIdx.x >> 5);
    int lane = threadIdx.x & 31;
    int n = gw >> 1, b = gw & 1;
    const float* base = agg2 + (size_t)n * JD2 + b * 256;   // [t][32]
    float acc = 0.0f;
#pragma unroll
    for (int t = 0; t < TT; ++t)
        acc += base[t * CHID + lane] * tw[t];
    acc *= W2[lane];
#pragma unroll
    for (int off = 16; off > 0; off >>= 1)
        acc += __shfl_xor(acc, off, 32);
    if (lane == 0) {
        float tws = 0.0f;
#pragma unroll
        for (int t = 0; t < TT; ++t) tws += tw[t];
        z[b * NN + n] = acc + b2[0] * tws + tb[0];
    }
}

// --------------------- 5) LN(n) -> scalar affine (tconv2) -> LN(n), exact.
// One block per batch; three strided passes over N=10000.
__global__ void ln_chain(const float* __restrict__ z,
                         const float* __restrict__ g1,
                         const float* __restrict__ bt1,
                         const float* __restrict__ tw2,
                         const float* __restrict__ tb2,
                         const float* __restrict__ g2,
                         const float* __restrict__ bt2,
                         float* __restrict__ u) {
    __shared__ float sm[1024];
    int b = blockIdx.x;
    int tid = threadIdx.x;
    const float* zb = z + b * NN;

    float s = 0.0f, ss = 0.0f;
    for (int n = tid; n < NN; n += blockDim.x) {
        float v = zb[n];
        s += v; ss += v * v;
    }
    float tot  = block_reduce_sum(s, sm);
    float tot2 = block_reduce_sum(ss, sm);
    float mu1  = tot / (float)NN;
    float var1 = tot2 / (float)NN - mu1 * mu1;
    float inv1 = rsqrtf(var1 + EPSLN);
    float w = tw2[0], c = tb2[0];

    s = 0.0f; ss = 0.0f;
    for (int n = tid; n < NN; n += blockDim.x) {
        float x = (zb[n] - mu1) * inv1 * g1[n] + bt1[n];
        float v = x * w + c;
        s += v; ss += v * v;
    }
    tot  = block_reduce_sum(s, sm);
    tot2 = block_reduce_sum(ss, sm);
    float mu2  = tot / (float)NN;
    float var2 = tot2 / (float)NN - mu2 * mu2;
    float inv2 = rsqrtf(var2 + EPSLN);

    for (int n = tid; n < NN; n += blockDim.x) {
        float x = (zb[n] - mu1) * inv1 * g1[n] + bt1[n];
        float v = x * w + c;
        u[b * NN + n] = (v - mu2) * inv2 * g2[n] + bt2[n];
    }
}

// ------------------------------------- 6) fc: 23 taps of a length-K dot, x2
__global__ void fc_conv(const float* __restrict__ u,
                        const float* __restrict__ fcw,
                        const float* __restrict__ fcb,
                        float* __restrict__ out) {
    __shared__ float sm[256];
    int idx = blockIdx.x;            // b*NCLASS + p
    int b = idx / NCLASS, p = idx % NCLASS;
    const float* ub = u + b * NN + p;
    float s = 0.0f;
    for (int k = threadIdx.x; k < KK; k += blockDim.x)
        s += ub[k] * fcw[k];
    float tot = block_reduce_sum(s, sm);
    if (threadIdx.x == 0) out[idx] = tot + fcb[0];
}

// ---------------------------------------------------------------- launcher
extern "C" void kernel_launch(void* const* d_in, const int* in_sizes, int n_in,
                              void* d_out, int out_size, void* d_ws, size_t ws_size,
                              hipStream_t stream) {
    (void)in_sizes; (void)n_in; (void)out_size; (void)ws_size;

    const float* feat = (const float*)d_in[0];
    const int*   src  = (const int*)d_in[1];
    const int*   dst  = (const int*)d_in[2];
    const float* W1   = (const float*)d_in[3];
    const float* b1   = (const float*)d_in[4];
    const float* W2   = (const float*)d_in[5];
    const float* b2   = (const float*)d_in[6];
    const float* tw1  = (const float*)d_in[7];
    const float* tb1  = (const float*)d_in[8];
    const float* g1   = (const float*)d_in[9];
    const float* bt1  = (const float*)d_in[10];
    const float* tw2  = (const float*)d_in[11];
    const float* tb2  = (const float*)d_in[12];
    const float* g2   = (const float*)d_in[13];
    const float* bt2  = (const float*)d_in[14];
    const float* fcw  = (const float*)d_in[15];
    const float* fcb  = (const float*)d_in[16];
    float* out = (float*)d_out;

    // Workspace (floats), with lifetime-based aliasing:
    //   [0 .. 2.56M)      xT   (dead after scatter_agg1)
    //   [2.56M .. 5.12M)  agg1 (dead after h1_wmma)
    //   [5.12M .. 10.24M) h1
    //   [0 .. 5.12M)      agg2 (reuses xT+agg1, zeroed after h1_wmma)
    //   [10.24M ..)       z, u
    float* ws   = (float*)d_ws;
    const size_t n_xT   = (size_t)NN * JD;    // 2,560,000
    const size_t n_agg1 = (size_t)NN * JD;    // 2,560,000
    const size_t n_h1   = (size_t)NN * JD2;   // 5,120,000
    const size_t n_agg2 = (size_t)NN * JD2;   // 5,120,000

    float* xT   = ws;
    float* agg1 = ws + n_xT;
    float* h1   = agg1 + n_agg1;
    float* agg2 = ws;                     // alias xT+agg1 (exactly 5.12M)
    float* z    = h1 + n_h1;
    float* u    = z + (size_t)BB * NN;

    // 0) node-major feature transpose (streaming, LDS-tiled)
    transpose_feat<<<dim3(NN / 16, JD / 16), 256, 0, stream>>>(feat, xT);

    // 1) layer-1 scatter
    zero_f32<<<((int)n_agg1 + 255) / 256, 256, 0, stream>>>(agg1, (int)n_agg1);
    scatter_agg1<<<EE / 4, 256, 0, stream>>>((const float4*)xT, src, dst, agg1);

    // 2) dense 16->32 projection + ReLU via WMMA (10000 wave-tiles)
    h1_wmma<<<1250, 256, 0, stream>>>(agg1, W1, b1, h1);

    // 3) layer-2 scatter (agg2 aliases xT/agg1 -> zero only now)
    zero_f32<<<((int)n_agg2 + 255) / 256, 256, 0, stream>>>(agg2, (int)n_agg2);
    scatter_agg2<<<EE / 2, 256, 0, stream>>>((const float4*)h1, src, dst, agg2);

    // 4) 32->1 projection fused with temporal conv (20000 (n,b) waves)
    h2_tconv1<<<2500, 256, 0, stream>>>(agg2, W2, b2, tw1, tb1, z);

    // 5) LayerNorm -> scalar affine -> LayerNorm
    ln_chain<<<BB, 1024, 0, stream>>>(z, g1, bt1, tw2, tb2, g2, bt2, u);

    // 6) fc over node dim -> [B, NCLASS]
    fc_conv<<<BB * NCLASS, 256, 0, stream>>>(u, fcw, fcb, out);
}